// MutualMultiAttentionHead_28784870818075
// MI455X (gfx1250) — compile-verified
//
#include <hip/hip_runtime.h>
#include <hip/hip_bf16.h>

// ---------------------------------------------------------------------------
// MutualMultiAttentionHead on MI455X (gfx1250, wave32)
//
// Sizes (compile-time constants from the reference):
//   H=4, N=1024, E=4096, LE=16384, FN=128, FE=64, FNO=128, FEO=64
//
// Strategy:
//  * All dense linears via V_WMMA_F32_16X16X4_F32 (fp32-exact tensor path).
//  * .reshape(H,-1,F) is a pure flat reinterpretation of the [M, H*F]
//    projection buffer -> index projections as flat[h*M*F + i*F + f].
//  * node_proj * A == node_proj (same scatter support), so the dense
//    [H,N,N]/[H,E,E] matmuls become sparse scatter-accumulates over the
//    edge lists using global_atomic_add_f32.
// ---------------------------------------------------------------------------

#define CH 4
#define CN 1024
#define CE 4096
#define CLE 16384
#define CFN 128
#define CFE 64
#define CFNO 128
#define CFEO 64

typedef __attribute__((ext_vector_type(2))) float v2f;
typedef __attribute__((ext_vector_type(8))) float v8f;

// ------------------------------ zero fill ----------------------------------
__global__ void k_zero(float* __restrict__ p, int n) {
    int i = blockIdx.x * blockDim.x + threadIdx.x;
    for (; i < n; i += gridDim.x * blockDim.x) p[i] = 0.0f;
}

// --------------------- generic fp32 WMMA linear ----------------------------
// Y[m,n] = sum_k X[m,k] * W[n,k] + b[n]   (torch Linear, W is [Nc,K])
// One wave per 16x16 output tile; K stepped by 4 via V_WMMA_F32_16X16X4_F32.
// Requires M%16==0, Nc%16==0, K%4==0 (all call sites satisfy this).
__global__ __launch_bounds__(128)
void k_linear_wmma(const float* __restrict__ X, const float* __restrict__ W,
                   const float* __restrict__ b, float* __restrict__ Y,
                   int M, int K, int Nc, int relu) {
    const int lane = threadIdx.x & 31;
    const int wave = threadIdx.x >> 5;
    const int tilesN = Nc >> 4;
    const int nt = (M >> 4) * tilesN;
    int tile = blockIdx.x * 4 + wave;          // wave-uniform
    if (tile >= nt) return;                    // uniform exit keeps EXEC all-1s

    const int tm = (tile / tilesN) << 4;
    const int tn = (tile % tilesN) << 4;
    const int half = lane >> 4;                // 0: K={0,1}, 1: K={2,3}
    const int l = lane & 15;

    const float* __restrict__ Arow = X + (size_t)(tm + l) * K + 2 * half;
    const float* __restrict__ Brow = W + (size_t)(tn + l) * K + 2 * half;

    v8f acc = {};
    for (int k = 0; k < K; k += 4) {
        float2 av = *(const float2*)(Arow + k);
        float2 bv = *(const float2*)(Brow + k);
        v2f a = {av.x, av.y};
        v2f bb = {bv.x, bv.y};
        // (neg_a, A, neg_b, B, c_mod, C, reuse_a, reuse_b)
        acc = __builtin_amdgcn_wmma_f32_16x16x4_f32(
            false, a, false, bb, (short)0, acc, false, false);
    }

    // C/D layout: lane l%16 -> col tn+l ; VGPR v -> row tm + v + 8*half
    const int col = tn + l;
    const float bias = b[col];
    float* __restrict__ yc = Y + (size_t)(tm + 8 * half) * Nc + col;
#pragma unroll
    for (int v = 0; v < 8; ++v) {
        float x = acc[v] + bias;
        if (relu) x = fmaxf(x, 0.0f);
        yc[(size_t)v * Nc] = x;
    }
}

// ------------------ narrow linear (K-projection, O=4) ----------------------
__global__ void k_small_linear(const float* __restrict__ X,
                               const float* __restrict__ W,
                               const float* __restrict__ b,
                               float* __restrict__ Y, int M, int K, int O) {
    int t = blockIdx.x * blockDim.x + threadIdx.x;
    if (t >= M * O) return;
    int m = t / O, o = t % O;
    const float4* x4 = (const float4*)(X + (size_t)m * K);
    const float4* w4 = (const float4*)(W + (size_t)o * K);
    float acc = 0.0f;
    for (int k = 0; k < (K >> 2); ++k) {
        float4 a = x4[k], w = w4[k];
        acc += a.x * w.x + a.y * w.y + a.z * w.z + a.w * w.w;
    }
    Y[t] = acc + b[o];
}

// ------------------- s[h,f] = sum_i Q[h,i,f] * Kv[h,i] ---------------------
// grid = H blocks, block = 256 threads (F columns x S strips)
__global__ void k_qk_reduce(const float* __restrict__ Q,
                            const float* __restrict__ Kv,
                            float* __restrict__ S, int M, int F) {
    __shared__ float sd[256];
    const int h = blockIdx.x;
    const int f = threadIdx.x % F;
    const int s = threadIdx.x / F;
    const int Sn = blockDim.x / F;
    float acc = 0.0f;
    const float* __restrict__ q = Q + (size_t)h * M * F;
    const float* __restrict__ kv = Kv + (size_t)h * M;
    for (int i = s; i < M; i += Sn) acc += q[(size_t)i * F + f] * kv[i];
    sd[threadIdx.x] = acc;
    __syncthreads();
    for (int st = Sn >> 1; st > 0; st >>= 1) {
        if (s < st) sd[threadIdx.x] += sd[threadIdx.x + st * F];
        __syncthreads();
    }
    if (s == 0) S[h * F + f] = sd[f];
}

// --------- logits[h,i] = Q[h,i,:].s[h,:]; softmax over i; -> out -----------
// grid = H blocks, block = 256 threads, R = M/256 rows per thread
__global__ void k_softmax_attn(const float* __restrict__ Q,
                               const float* __restrict__ Svec,
                               float* __restrict__ out, int M, int F) {
    __shared__ float sd[256];
    const int h = blockIdx.x;
    const int tid = threadIdx.x;
    const int T = 256;
    const int R = M / T;
    float logit[16];

    const float4* s4 = (const float4*)(Svec + (size_t)h * F);
    float lmax = -3.4e38f;
    for (int r = 0; r < R; ++r) {
        int i = tid + r * T;
        const float4* q4 = (const float4*)(Q + (size_t)h * M * F + (size_t)i * F);
        float acc = 0.0f;
        for (int f = 0; f < (F >> 2); ++f) {
            float4 a = q4[f], s = s4[f];
            acc += a.x * s.x + a.y * s.y + a.z * s.z + a.w * s.w;
        }
        logit[r] = acc;
        lmax = fmaxf(lmax, acc);
    }
    sd[tid] = lmax;
    __syncthreads();
    for (int st = T >> 1; st > 0; st >>= 1) {
        if (tid < st) sd[tid] = fmaxf(sd[tid], sd[tid + st]);
        __syncthreads();
    }
    const float gmax = sd[0];
    __syncthreads();

    float lsum = 0.0f;
    for (int r = 0; r < R; ++r) {
        logit[r] = __expf(logit[r] - gmax);
        lsum += logit[r];
    }
    sd[tid] = lsum;
    __syncthreads();
    for (int st = T >> 1; st > 0; st >>= 1) {
        if (tid < st) sd[tid] += sd[tid + st];
        __syncthreads();
    }
    const float inv = 1.0f / sd[0];
    for (int r = 0; r < R; ++r) out[(size_t)h * M + tid + r * T] = logit[r] * inv;
}

// --- node aggregation: agg[h, src[e], :] += edge_sa[h,e] * nv[h, dst[e], :] -
// one wave per (h,e); FN=128 floats -> 4 atomics per lane
__global__ void k_node_scatter(const int* __restrict__ src,
                               const int* __restrict__ dst,
                               const float* __restrict__ esa,
                               const float* __restrict__ nv,
                               float* __restrict__ agg) {
    int widx = (blockIdx.x * blockDim.x + threadIdx.x) >> 5;
    int lane = threadIdx.x & 31;
    if (widx >= CH * CE) return;
    int h = widx / CE, e = widx % CE;
    int r = src[e], d = dst[e];
    float w = esa[(size_t)h * CE + e];
    const float* __restrict__ v = nv + (size_t)h * CN * CFN + (size_t)d * CFN;
    float* __restrict__ o = agg + (size_t)h * CN * CFN + (size_t)r * CFN;
#pragma unroll
    for (int f = lane; f < CFN; f += 32) atomicAdd(&o[f], w * v[f]);
}

// --- edge aggregation: agg[h, lg_src[j], :] += node_sa[h, dst[lg_src[j]]] *
//                                               ev[h, lg_dst[j], :]
__global__ void k_edge_scatter(const int* __restrict__ lg_src,
                               const int* __restrict__ lg_dst,
                               const int* __restrict__ dst,
                               const float* __restrict__ nsa,
                               const float* __restrict__ ev,
                               float* __restrict__ agg) {
    int widx = (blockIdx.x * blockDim.x + threadIdx.x) >> 5;
    int lane = threadIdx.x & 31;
    if (widx >= CH * CLE) return;
    int h = widx / CLE, j = widx % CLE;
    int er = lg_src[j], ec = lg_dst[j];
    int conn = dst[er];
    float w = nsa[(size_t)h * CN + conn];
    const float* __restrict__ v = ev + (size_t)h * CE * CFE + (size_t)ec * CFE;
    float* __restrict__ o = agg + (size_t)h * CE * CFE + (size_t)er * CFE;
#pragma unroll
    for (int f = lane; f < CFE; f += 32) atomicAdd(&o[f], w * v[f]);
}

// ---------------------------------------------------------------------------
extern "C" void kernel_launch(void* const* d_in, const int* in_sizes, int n_in,
                              void* d_out, int out_size, void* d_ws, size_t ws_size,
                              hipStream_t stream) {
    const float* node_in = (const float*)d_in[0];
    const float* edge_in = (const float*)d_in[1];
    const int* src    = (const int*)d_in[2];
    const int* dst    = (const int*)d_in[3];
    const int* lg_src = (const int*)d_in[4];
    const int* lg_dst = (const int*)d_in[5];
    const float* nqW = (const float*)d_in[6],  *nqb = (const float*)d_in[7];
    const float* nkW = (const float*)d_in[8],  *nkb = (const float*)d_in[9];
    const float* nvW = (const float*)d_in[10], *nvb = (const float*)d_in[11];
    const float* eqW = (const float*)d_in[12], *eqb = (const float*)d_in[13];
    const float* ekW = (const float*)d_in[14], *ekb = (const float*)d_in[15];
    const float* evW = (const float*)d_in[16], *evb = (const float*)d_in[17];
    const float* ncW = (const float*)d_in[18], *ncb = (const float*)d_in[19];
    const float* ecW = (const float*)d_in[20], *ecb = (const float*)d_in[21];

    float* ws   = (float*)d_ws;
    float* nqp  = ws;                    // N * H*FN      = 524288
    float* nvp  = nqp  + 524288;         // 524288
    float* nkp  = nvp  + 524288;         // N * H         = 4096
    float* eqp  = nkp  + 4096;           // E * H*FE      = 1048576
    float* evp  = eqp  + 1048576;        // 1048576
    float* ekp  = evp  + 1048576;        // E * H         = 16384
    float* sn   = ekp  + 16384;          // H * FN        = 512
    float* se   = sn   + 512;            // H * FE        = 256
    float* nsa  = se   + 256;            // H * N         = 4096
    float* esa  = nsa  + 4096;           // H * E         = 16384
    float* nagg = esa  + 16384;          // H * N * FN    = 524288
    float* eagg = nagg + 524288;         // H * E * FE    = 1048576

    // zero both aggregation buffers (contiguous) each call (determinism)
    k_zero<<<512, 256, 0, stream>>>(nagg, 524288 + 1048576);

    // --- projections (WMMA fp32) ---
    // nq / nv: [1024,128] @ [128,512]^T -> 2048 tiles
    k_linear_wmma<<<2048 / 4, 128, 0, stream>>>(node_in, nqW, nqb, nqp, CN, CFN, CH * CFN, 0);
    k_linear_wmma<<<2048 / 4, 128, 0, stream>>>(node_in, nvW, nvb, nvp, CN, CFN, CH * CFN, 0);
    // eq / ev: [4096,64] @ [64,256]^T -> 4096 tiles
    k_linear_wmma<<<4096 / 4, 128, 0, stream>>>(edge_in, eqW, eqb, eqp, CE, CFE, CH * CFE, 0);
    k_linear_wmma<<<4096 / 4, 128, 0, stream>>>(edge_in, evW, evb, evp, CE, CFE, CH * CFE, 0);
    // nk: [1024,128] -> [1024,4] ; ek: [4096,64] -> [4096,4]
    k_small_linear<<<(CN * CH + 255) / 256, 256, 0, stream>>>(node_in, nkW, nkb, nkp, CN, CFN, CH);
    k_small_linear<<<(CE * CH + 255) / 256, 256, 0, stream>>>(edge_in, ekW, ekb, ekp, CE, CFE, CH);

    // --- rank-1 attention:  s = Q^T k ;  softmax(Q s) over rows ---
    k_qk_reduce<<<CH, 256, 0, stream>>>(nqp, nkp, sn, CN, CFN);
    k_qk_reduce<<<CH, 256, 0, stream>>>(eqp, ekp, se, CE, CFE);
    k_softmax_attn<<<CH, 256, 0, stream>>>(nqp, sn, nsa, CN, CFN);
    k_softmax_attn<<<CH, 256, 0, stream>>>(eqp, se, esa, CE, CFE);

    // --- sparse scatter-aggregation (replaces dense (proj*A) @ V) ---
    // node: H*E waves ; edge: H*LE waves ; 8 waves (256 thr) per block
    k_node_scatter<<<(CH * CE) / 8, 256, 0, stream>>>(src, dst, esa, nvp, nagg);
    k_edge_scatter<<<(CH * CLE) / 8, 256, 0, stream>>>(lg_src, lg_dst, dst, nsa, evp, eagg);

    // --- output linears + relu (WMMA fp32) ---
    float* node_out = (float*)d_out;                 // [H,N,FNO] = 524288
    float* edge_out = node_out + CH * CN * CFNO;     // [H,E,FEO] = 1048576
    // [4096,128] @ [128,128]^T -> 2048 tiles
    k_linear_wmma<<<2048 / 4, 128, 0, stream>>>(nagg, ncW, ncb, node_out, CH * CN, CFN, CFNO, 1);
    // [16384,64] @ [64,64]^T -> 4096 tiles
    k_linear_wmma<<<4096 / 4, 128, 0, stream>>>(eagg, ecW, ecb, edge_out, CH * CE, CFE, CFEO, 1);
}